// GIN_77661598646384
// MI455X (gfx1250) — compile-verified
//
#include <hip/hip_runtime.h>

#define NNODES 100000
#define NEDGES 3200000
#define DIN 128
#define HID 128
#define NLAYERS 4
#define NGRAPH 1000
#define NCLS 10
#define BN_EPS 1e-5f

typedef __attribute__((ext_vector_type(2))) float v2f;
typedef __attribute__((ext_vector_type(8))) float v8f;

__device__ __forceinline__ void atomAddF(float* p, float v) {
  __hip_atomic_fetch_add(p, v, __ATOMIC_RELAXED, __HIP_MEMORY_SCOPE_AGENT);
}

// ---------------------------------------------------------------- utilities
__global__ void zero_f32(float* __restrict__ p, int n) {
  int i = blockIdx.x * blockDim.x + threadIdx.x;
  if (i < n) p[i] = 0.0f;
}

// z = (1 + eps[l]) * h   (self term of GINConv)
__global__ void zinit_f4(const float4* __restrict__ h, const float* __restrict__ eps, int l,
                         float4* __restrict__ z, int n4) {
  int i = blockIdx.x * blockDim.x + threadIdx.x;
  if (i >= n4) return;
  float s = 1.0f + eps[l];
  float4 v = h[i];
  v.x *= s; v.y *= s; v.z *= s; v.w *= s;
  z[i] = v;
}

// z[dst] += h[src] over all edges; one lane per (edge, 4-float chunk)
__global__ void edge_agg(const float* __restrict__ h, const int* __restrict__ ei,
                         float* __restrict__ z) {
  size_t tid = (size_t)blockIdx.x * blockDim.x + threadIdx.x;
  int e = (int)(tid >> 5);
  if (e >= NEDGES) return;
  int c = ((int)tid & 31) << 2;
  int s = ei[e];
  int d = ei[NEDGES + e];
  const float4 v = *(const float4*)(h + (size_t)s * HID + c);
  float* zp = z + (size_t)d * HID + c;
  atomAddF(zp + 0, v.x);
  atomAddF(zp + 1, v.y);
  atomAddF(zp + 2, v.z);
  atomAddF(zp + 3, v.w);
}

// ------------------------------------------------------------- WMMA GEMM
// out = preop(A) @ B + bias, with per-column sum / sum-of-squares accumulated
// for the following BatchNorm.  preop(x) = relu(x*scale[k] + shift[k]) if
// PRE, else identity.  Block: 256 threads = 8 waves, 128 rows x (CT*16) cols.
// B fragments staged in LDS pre-swizzled in WMMA operand order; BN affine
// staged in LDS as float2 pairs.  Inner loop is branchless (EXEC all-ones at
// every V_WMMA) and the A fetch is software-pipelined one step ahead so its
// latency is hidden under the WMMAs of the current step.
template <int CT, bool PRE>
__global__ __launch_bounds__(256) void gemm_wmma(
    const float* __restrict__ A, const float* __restrict__ B, const float* __restrict__ bias,
    const float* __restrict__ preScale, const float* __restrict__ preShift,
    float* __restrict__ out, float* __restrict__ colSum, float* __restrict__ colSqSum,
    int nrows, int kdim, int ncols) {
  extern __shared__ char smem_raw[];
  float* lds = (float*)smem_raw;  // kdim * CT*16 floats of B fragments

  const int tid = threadIdx.x;
  const int lane = tid & 31;
  const int wave = tid >> 5;
  const int colBase = blockIdx.y * (CT * 16);
  const int rowBase = blockIdx.x * 128 + wave * 16;

  // Stage B in WMMA-fragment order: element i = ((t*CT + c)*32 + lane)*2 + j
  // holds B[4t + (lane>>4)*2 + j][colBase + c*16 + (lane&15)].
  const int totalElems = kdim * CT * 16;
  constexpr int SH = (CT == 4) ? 8 : 7;  // log2(CT*64)
  for (int i = tid; i < totalElems; i += 256) {
    int j = i & 1;
    int ln = (i >> 1) & 31;
    int c = (i >> 6) & (CT - 1);
    int t = i >> SH;
    int k = (t << 2) + ((ln >> 4) << 1) + j;
    int col = colBase + c * 16 + (ln & 15);
    lds[i] = B[(size_t)k * ncols + col];
  }
  float2* ldsP = (float2*)(lds + totalElems);  // {scale,shift}[kdim]
  if (PRE) {
    for (int i = tid; i < kdim; i += 256) ldsP[i] = make_float2(preScale[i], preShift[i]);
  }
  __syncthreads();

  const int mrow = rowBase + (lane & 15);
  const int kk = (lane >> 4) << 1;  // K sub-offset of this half-wave
  const float rowMask = (mrow < nrows) ? 1.0f : 0.0f;
  const int mrowC = min(mrow, nrows - 1);  // clamp: loads always in-bounds

  v8f acc[CT] = {};
  const v2f* bfrag = (const v2f*)lds;
  const float* aptr = A + (size_t)mrowC * kdim + kk;

  // software pipeline: A fragment for step k is fetched during step k-4.
  float c0 = aptr[0];
  float c1 = aptr[1];
  for (int k = 0; k < kdim; k += 4) {
    // NOTE: final iteration reads <=16B past the row; all A operands are
    // interior workspace buffers, so this is always mapped memory.
    const float n0 = aptr[k + 4];
    const float n1 = aptr[k + 5];
    __builtin_prefetch(aptr + k + 64, 0, 1);  // global_prefetch_b8, 2 lines ahead
    float a0 = c0, a1 = c1;
    if (PRE) {
      const float2 p0 = ldsP[k + kk];  // broadcast ds load (one addr/half-wave)
      const float2 p1 = ldsP[k + kk + 1];
      a0 = fmaxf(fmaf(a0, p0.x, p0.y), 0.0f);
      a1 = fmaxf(fmaf(a1, p1.x, p1.y), 0.0f);
    }
    a0 *= rowMask;  // zero padded rows without touching EXEC
    a1 *= rowMask;
    v2f a;
    a.x = a0; a.y = a1;
    const int t = k >> 2;
#pragma unroll
    for (int c = 0; c < CT; ++c) {
      v2f b = bfrag[(t * CT + c) * 32 + lane];
      acc[c] = __builtin_amdgcn_wmma_f32_16x16x4_f32(false, a, false, b, (short)0, acc[c],
                                                     false, false);
    }
    c0 = n0;
    c1 = n1;
  }

  // Epilogue: bias, store, per-column batch statistics (LDS reduction).
  __syncthreads();  // B staging no longer needed; reuse LDS
  float* redS = lds;                // [8][CT][16]
  float* redQ = lds + 8 * CT * 16;  // [8][CT][16]

  float psum[CT], psq[CT];
#pragma unroll
  for (int c = 0; c < CT; ++c) {
    const int col = colBase + c * 16 + (lane & 15);
    const float bv = bias[col];
    float s = 0.0f, q = 0.0f;
#pragma unroll
    for (int r = 0; r < 8; ++r) {
      const int row = rowBase + ((lane >> 4) << 3) + r;
      if (row < nrows) {
        const float y = acc[c][r] + bv;
        out[(size_t)row * ncols + col] = y;
        s += y;
        q += y * y;
      }
    }
    s += __shfl_xor(s, 16);  // lanes L and L+16 hold the same column
    q += __shfl_xor(q, 16);
    psum[c] = s;
    psq[c] = q;
  }
  if (lane < 16) {
#pragma unroll
    for (int c = 0; c < CT; ++c) {
      redS[(wave * CT + c) * 16 + lane] = psum[c];
      redQ[(wave * CT + c) * 16 + lane] = psq[c];
    }
  }
  __syncthreads();
  if (tid < CT * 16) {
    const int c = tid >> 4, cl = tid & 15;
    float s = 0.0f, q = 0.0f;
#pragma unroll
    for (int w = 0; w < 8; ++w) {
      s += redS[(w * CT + c) * 16 + cl];
      q += redQ[(w * CT + c) * 16 + cl];
    }
    atomAddF(colSum + colBase + tid, s);
    atomAddF(colSqSum + colBase + tid, q);
  }
}

// stats -> per-column affine (scale, shift) implementing BN in one fma
__global__ void finalize_stats(const float* __restrict__ sum, const float* __restrict__ sq,
                               const float* __restrict__ gamma, const float* __restrict__ beta,
                               float* __restrict__ scale, float* __restrict__ shift, int ncols,
                               float invCount) {
  int c = blockIdx.x * blockDim.x + threadIdx.x;
  if (c >= ncols) return;
  float m = sum[c] * invCount;
  float v = sq[c] * invCount - m * m;
  float sc = gamma[c] * rsqrtf(v + BN_EPS);
  scale[c] = sc;
  shift[c] = fmaf(-m, sc, beta[c]);
}

// h = relu(t * scale + shift), 128-wide rows
__global__ void bn_relu_f4(const float4* __restrict__ t, const float* __restrict__ scale,
                           const float* __restrict__ shift, float4* __restrict__ h, int n4) {
  int i = blockIdx.x * blockDim.x + threadIdx.x;
  if (i >= n4) return;
  int cb = (i & 31) << 2;
  float4 v = t[i];
  v.x = fmaxf(fmaf(v.x, scale[cb + 0], shift[cb + 0]), 0.0f);
  v.y = fmaxf(fmaf(v.y, scale[cb + 1], shift[cb + 1]), 0.0f);
  v.z = fmaxf(fmaf(v.z, scale[cb + 2], shift[cb + 2]), 0.0f);
  v.w = fmaxf(fmaf(v.w, scale[cb + 3], shift[cb + 3]), 0.0f);
  h[i] = v;
}

// global_add_pool: ge[batch[n]] += h[n]
__global__ void pool_add(const float* __restrict__ h, const int* __restrict__ batch,
                         float* __restrict__ ge) {
  size_t tid = (size_t)blockIdx.x * blockDim.x + threadIdx.x;
  int n = (int)(tid >> 5);
  if (n >= NNODES) return;
  int c = ((int)tid & 31) << 2;
  int g = batch[n];
  const float4 v = *(const float4*)(h + (size_t)n * HID + c);
  float* gp = ge + (size_t)g * HID + c;
  atomAddF(gp + 0, v.x);
  atomAddF(gp + 1, v.y);
  atomAddF(gp + 2, v.z);
  atomAddF(gp + 3, v.w);
}

// out = relu(bn(tg)) @ Wm2 + bm2   (tiny: 1000 x 10, K=128)
__global__ void head_gemm(const float* __restrict__ tg, const float* __restrict__ scale,
                          const float* __restrict__ shift, const float* __restrict__ W,
                          const float* __restrict__ b, float* __restrict__ out) {
  int i = blockIdx.x * blockDim.x + threadIdx.x;
  if (i >= NGRAPH * NCLS) return;
  int g = i / NCLS, c = i % NCLS;
  const float* row = tg + (size_t)g * HID;
  float acc = b[c];
  for (int k = 0; k < HID; ++k) {
    float a = fmaxf(fmaf(row[k], scale[k], shift[k]), 0.0f);
    acc = fmaf(a, W[k * NCLS + c], acc);
  }
  out[i] = acc;
}

// ------------------------------------------------------------------ driver
extern "C" void kernel_launch(void* const* d_in, const int* in_sizes, int n_in, void* d_out,
                              int out_size, void* d_ws, size_t ws_size, hipStream_t stream) {
  const float* x = (const float*)d_in[0];
  const int* ei = (const int*)d_in[1];
  const int* bat = (const int*)d_in[2];
  const float* eps = (const float*)d_in[3];
  const float* W1 = (const float*)d_in[4];
  const float* b1 = (const float*)d_in[5];
  const float* g1 = (const float*)d_in[6];
  const float* bt1 = (const float*)d_in[7];
  const float* W2 = (const float*)d_in[8];
  const float* b2 = (const float*)d_in[9];
  const float* g2 = (const float*)d_in[10];
  const float* bt2 = (const float*)d_in[11];
  const float* Wm1 = (const float*)d_in[12];
  const float* bm1 = (const float*)d_in[13];
  const float* gm = (const float*)d_in[14];
  const float* btm = (const float*)d_in[15];
  const float* Wm2 = (const float*)d_in[16];
  const float* bm2 = (const float*)d_in[17];
  float* out = (float*)d_out;

  float* ws = (float*)d_ws;
  float* hbuf = ws;                            // N*128
  float* zbuf = hbuf + (size_t)NNODES * HID;   // N*128 (also reused as t2)
  float* t1 = zbuf + (size_t)NNODES * HID;     // N*256
  float* sum = t1 + (size_t)NNODES * 2 * HID;  // 256
  float* sq = sum + 256;                       // 256
  float* scl = sq + 256;                       // 256
  float* shf = scl + 256;                      // 256
  float* ge = shf + 256;                       // G*128
  float* tg = ge + (size_t)NGRAPH * HID;       // G*128

  const int n4 = NNODES * HID / 4;
  const int edgeBlocks = (int)(((size_t)NEDGES * 32 + 255) / 256);
  const int gRows = (NNODES + 127) / 128;

  const size_t lds1 = 128 * 64 * sizeof(float);                         // CT=4, K=128
  const size_t lds2 = 256 * 32 * sizeof(float) + 256 * sizeof(float2);  // CT=2, K=256, +BN

  const float* hcur = x;
  for (int l = 0; l < NLAYERS; ++l) {
    zinit_f4<<<(n4 + 255) / 256, 256, 0, stream>>>((const float4*)hcur, eps, l, (float4*)zbuf,
                                                   n4);
    edge_agg<<<edgeBlocks, 256, 0, stream>>>(hcur, ei, zbuf);

    zero_f32<<<2, 256, 0, stream>>>(sum, 512);
    gemm_wmma<4, false><<<dim3(gRows, 4), 256, lds1, stream>>>(
        zbuf, W1 + (size_t)l * DIN * 2 * HID, b1 + l * 2 * HID, nullptr, nullptr, t1, sum, sq,
        NNODES, DIN, 2 * HID);
    finalize_stats<<<1, 256, 0, stream>>>(sum, sq, g1 + l * 2 * HID, bt1 + l * 2 * HID, scl,
                                          shf, 2 * HID, 1.0f / NNODES);

    zero_f32<<<2, 256, 0, stream>>>(sum, 512);
    gemm_wmma<2, true><<<dim3(gRows, 4), 256, lds2, stream>>>(
        t1, W2 + (size_t)l * 2 * HID * HID, b2 + l * HID, scl, shf, zbuf, sum, sq, NNODES,
        2 * HID, HID);
    finalize_stats<<<1, 256, 0, stream>>>(sum, sq, g2 + l * HID, bt2 + l * HID, scl, shf, HID,
                                          1.0f / NNODES);

    bn_relu_f4<<<(n4 + 255) / 256, 256, 0, stream>>>((const float4*)zbuf, scl, shf,
                                                     (float4*)hbuf, n4);
    hcur = hbuf;
  }

  zero_f32<<<(NGRAPH * HID + 255) / 256, 256, 0, stream>>>(ge, NGRAPH * HID);
  pool_add<<<(int)(((size_t)NNODES * 32 + 255) / 256), 256, 0, stream>>>(hcur, bat, ge);

  zero_f32<<<2, 256, 0, stream>>>(sum, 512);
  gemm_wmma<4, false><<<dim3((NGRAPH + 127) / 128, 2), 256, lds1, stream>>>(
      ge, Wm1, bm1, nullptr, nullptr, tg, sum, sq, NGRAPH, HID, HID);
  finalize_stats<<<1, 256, 0, stream>>>(sum, sq, gm, btm, scl, shf, HID, 1.0f / NGRAPH);

  head_gemm<<<(NGRAPH * NCLS + 255) / 256, 256, 0, stream>>>(tg, scl, shf, Wm2, bm2, out);
}